// _LayerGCN_51668456571008
// MI455X (gfx1250) — compile-verified
//
#include <hip/hip_runtime.h>
#include <hip/hip_bf16.h>

typedef __attribute__((ext_vector_type(2))) float v2f;
typedef __attribute__((ext_vector_type(8))) float v8f;

#define EMBED_DIM 128
#define EPS_COS 1e-8f

// ---------------------------------------------------------------------------
// Build ego = concat(user_emb, item_emb); x = ego; acc (= d_out) = 0.
// ---------------------------------------------------------------------------
__global__ void lgcn_init(const float* __restrict__ uemb,
                          const float* __restrict__ iemb,
                          float* __restrict__ ego,
                          float* __restrict__ x,
                          float* __restrict__ acc,
                          long long n_user_elems, long long total_elems) {
  long long i = (long long)blockIdx.x * blockDim.x + threadIdx.x;
  long long stride = (long long)gridDim.x * blockDim.x;
  for (; i < total_elems; i += stride) {
    float v = (i < n_user_elems) ? uemb[i] : iemb[i - n_user_elems];
    ego[i] = v;
    x[i] = v;
    acc[i] = 0.0f;
  }
}

// ---------------------------------------------------------------------------
// Per-node 1/max(||ego||, eps).  One wave32 per node; lane owns a float4.
// ---------------------------------------------------------------------------
__global__ void lgcn_ego_norm(const float* __restrict__ ego,
                              float* __restrict__ inv_norm, int n_nodes) {
  int lane = threadIdx.x & 31;
  int node = (blockIdx.x * blockDim.x + threadIdx.x) >> 5;
  if (node >= n_nodes) return;
  const float4 v = *(const float4*)(ego + (size_t)node * EMBED_DIM + lane * 4);
  float s = v.x * v.x + v.y * v.y + v.z * v.z + v.w * v.w;
#pragma unroll
  for (int off = 16; off >= 1; off >>= 1) s += __shfl_xor(s, off, 32);
  if (lane == 0) inv_norm[node] = 1.0f / fmaxf(sqrtf(s), EPS_COS);
}

// ---------------------------------------------------------------------------
// SpMM scatter: y[r] += v * x[c].  One wave32 per edge, lane owns float4.
// x (76.8 MB) and y fit in the 192 MB L2, so gathers + f32 atomics stay
// on-chip; HBM only streams the 120 MB edge list per layer.
// ---------------------------------------------------------------------------
__global__ void lgcn_spmm(const int* __restrict__ rows,
                          const int* __restrict__ cols,
                          const float* __restrict__ vals,
                          const float* __restrict__ x,
                          float* __restrict__ y, int nnz) {
  int lane = threadIdx.x & 31;
  long long wave = (long long)((blockIdx.x * blockDim.x + threadIdx.x) >> 5);
  long long nwaves = ((long long)gridDim.x * blockDim.x) >> 5;
  for (long long e = wave; e < nnz; e += nwaves) {
    // prefetch next edge record this wave will touch (global_prefetch_b8)
    __builtin_prefetch(&rows[e + nwaves], 0, 1);
    __builtin_prefetch(&cols[e + nwaves], 0, 1);
    __builtin_prefetch(&vals[e + nwaves], 0, 1);
    int r = rows[e];
    int c = cols[e];
    float v = vals[e];
    const float4 xv = *(const float4*)(x + (size_t)c * EMBED_DIM + lane * 4);
    float* dst = y + (size_t)r * EMBED_DIM + lane * 4;
    atomicAdd(dst + 0, xv.x * v);
    atomicAdd(dst + 1, xv.y * v);
    atomicAdd(dst + 2, xv.z * v);
    atomicAdd(dst + 3, xv.w * v);
  }
}

// ---------------------------------------------------------------------------
// Cosine reweight + accumulate.  One wave32 per 16-node tile.
// Uses V_WMMA_F32_16X16X4_F32 to build diag(Y*Ego^T) and diag(Y*Y^T) in
// full fp32.  A-fragment for a 16x4 f32 tile: lane L holds
// (M = L&15, K = 4*kk + 2*(L>>4) + {0,1}); the B-fragment of the transposed
// operand has the identical per-lane values, so wmma(a,a) = Y*Y^T.
// ---------------------------------------------------------------------------
__global__ void lgcn_weight_scale(const float* __restrict__ y,
                                  const float* __restrict__ ego,
                                  const float* __restrict__ ego_inv_norm,
                                  float* __restrict__ xnext,
                                  float* __restrict__ acc, int n_nodes) {
  __shared__ float s_w[8][16];  // per-wave weight broadcast
  const int lane = threadIdx.x & 31;
  const int wv = threadIdx.x >> 5;
  const int tile = (blockIdx.x * blockDim.x + threadIdx.x) >> 5;
  const int node0 = tile * 16;
  const bool active = (node0 < n_nodes);
  const int m = lane & 15;

  if (active) {
    if (node0 + 16 <= n_nodes) {
      // ---- full tile: WMMA path (EXEC all ones inside this uniform branch)
      const int koff = (lane >> 4) << 1;  // 0 or 2
      const float* yrow = y + (size_t)(node0 + m) * EMBED_DIM;
      const float* erow = ego + (size_t)(node0 + m) * EMBED_DIM;
      v8f cNum = {};
      v8f cYY = {};
#pragma unroll
      for (int kk = 0; kk < EMBED_DIM / 4; ++kk) {
        const int kb = kk * 4 + koff;
        v2f a;
        a.x = yrow[kb];
        a.y = yrow[kb + 1];
        v2f b;
        b.x = erow[kb];
        b.y = erow[kb + 1];
        cNum = __builtin_amdgcn_wmma_f32_16x16x4_f32(false, a, false, b,
                                                     (short)0, cNum, false, false);
        cYY = __builtin_amdgcn_wmma_f32_16x16x4_f32(false, a, false, a,
                                                    (short)0, cYY, false, false);
      }
      // Diagonal (M==N): node n<8 lives in lane n, vgpr n;
      // node n>=8 lives in lane n+16, vgpr n-8.
      const bool holder = (m < 8) ? (lane == m) : (lane == m + 16);
      const int vi = (m < 8) ? m : (m - 8);
      float num = 0.0f, yy = 0.0f;
#pragma unroll
      for (int i = 0; i < 8; ++i) {
        if (vi == i) { num = cNum[i]; yy = cYY[i]; }
      }
      if (holder) {
        float inv_yn = 1.0f / fmaxf(sqrtf(fmaxf(yy, 0.0f)), EPS_COS);
        s_w[wv][m] = num * inv_yn * ego_inv_norm[node0 + m];
      }
    } else {
      // ---- tail tile (rem < 16): scalar per-lane fallback
      int rem = n_nodes - node0;
      if (lane < rem) {
        int node = node0 + lane;
        const float* yrow = y + (size_t)node * EMBED_DIM;
        const float* erow = ego + (size_t)node * EMBED_DIM;
        float num = 0.0f, yy = 0.0f;
        for (int k = 0; k < EMBED_DIM; ++k) {
          float yv = yrow[k];
          num += yv * erow[k];
          yy += yv * yv;
        }
        float inv_yn = 1.0f / fmaxf(sqrtf(yy), EPS_COS);
        s_w[wv][lane] = num * inv_yn * ego_inv_norm[node];
      }
    }
  }
  __syncthreads();
  if (!active) return;

  const int rem = min(16, n_nodes - node0);
  for (int j = 0; j < rem; ++j) {
    const float w = s_w[wv][j];
    const size_t base = (size_t)(node0 + j) * EMBED_DIM + lane * 4;
    const float4 v = *(const float4*)(y + base);
    float4 sv;
    sv.x = v.x * w;
    sv.y = v.y * w;
    sv.z = v.z * w;
    sv.w = v.w * w;
    *(float4*)(xnext + base) = sv;  // input to next layer's SpMM
    float4 a = *(const float4*)(acc + base);
    a.x += sv.x;
    a.y += sv.y;
    a.z += sv.z;
    a.w += sv.w;
    *(float4*)(acc + base) = a;
  }
}

// ---------------------------------------------------------------------------
// Host-side orchestration (graph-capture safe: only kernels + memsetAsync).
// ---------------------------------------------------------------------------
extern "C" void kernel_launch(void* const* d_in, const int* in_sizes, int n_in,
                              void* d_out, int out_size, void* d_ws,
                              size_t ws_size, hipStream_t stream) {
  const float* uemb = (const float*)d_in[0];
  const float* iemb = (const float*)d_in[1];
  const int* rows = (const int*)d_in[2];
  const int* cols = (const int*)d_in[3];
  const float* vals = (const float*)d_in[4];

  const long long nue = in_sizes[0];          // n_users * 128
  const long long nie = in_sizes[1];          // n_items * 128
  const int nnz = in_sizes[2];                // 2 * NNZ directed edges
  const long long total = nue + nie;          // n_nodes * 128
  const int n_nodes = (int)(total / EMBED_DIM);

  // Workspace layout: ego | x | y | inv_norm  (3*76.8 MB + 0.6 MB)
  float* ego = (float*)d_ws;
  float* x = ego + total;
  float* yb = x + total;
  float* inv_norm = yb + total;
  float* acc = (float*)d_out;

  lgcn_init<<<4096, 256, 0, stream>>>(uemb, iemb, ego, x, acc, nue, total);

  {
    int blocks = (n_nodes + 7) / 8;  // 8 waves (nodes) per 256-thread block
    lgcn_ego_norm<<<blocks, 256, 0, stream>>>(ego, inv_norm, n_nodes);
  }

  const int tiles = (n_nodes + 15) / 16;
  const int p2_blocks = (tiles + 7) / 8;  // 8 wave-tiles per block

  for (int layer = 0; layer < 4; ++layer) {
    hipMemsetAsync(yb, 0, (size_t)total * sizeof(float), stream);
    lgcn_spmm<<<32768, 256, 0, stream>>>(rows, cols, vals, x, yb, nnz);
    lgcn_weight_scale<<<p2_blocks, 256, 0, stream>>>(yb, ego, inv_norm, x, acc,
                                                     n_nodes);
  }
}